// FDTD2DLayer_17978733101078
// MI455X (gfx1250) — compile-verified
//
#include <hip/hip_runtime.h>
#include <math.h>

typedef __attribute__((ext_vector_type(2))) float v2f;
typedef __attribute__((ext_vector_type(8))) float v8f;
typedef __attribute__((ext_vector_type(4))) unsigned int v4u;
typedef __attribute__((ext_vector_type(4))) int v4i;
typedef __attribute__((ext_vector_type(8))) int v8i;

#define GS 96
#define G  (GS * GS)   // 9216
#define H  512
#define T  2048

#define USE_TDM 1      // flip to 0 to fall back to direct-global loads

// ---------------------------------------------------------------------------
// TDM: issue a 2D tile load (f32 elements) global -> LDS via the Tensor Data
// Mover. Descriptor bitfields per CDNA5 ISA ch.10.8 (D# groups 0/1).
// rows x 32 dwords per row, LDS rows padded to 34 dwords via TDM pad engine.
// ---------------------------------------------------------------------------
#if USE_TDM
__device__ __forceinline__ void tdm_load_tile_f32(
    unsigned lds_byte_addr,           // LDS byte offset of tile start
    const float* gsrc,                // global address of tile start
    unsigned tensor_d0,               // row length of source tensor (elements)
    unsigned long long row_stride,    // stride between rows (elements)
    unsigned tile_rows)               // tile_dim1 (rows); tile_dim0 fixed = 32
{
    const unsigned long long ga = (unsigned long long)gsrc;
    v4u g0;
    g0.x = 1u;                                           // count=1 (valid), user mode
    g0.y = lds_byte_addr;                                // [63:32] lds_addr
    g0.z = (unsigned)(ga & 0xFFFFFFFFu);                 // [95:64] global_addr lo
    g0.w = (unsigned)((ga >> 32) & 0x1FFFFFFu)           // [120:96] global_addr hi
         | (2u << 30);                                   // [127:126] type=2 (image)

    v8i g1;
    // dword0: wg_mask=0 | data_size=2 (4B) | pad_enable | pad_interval=4 (32 dw)
    //         | pad_amount code 1 (2 dwords)
    g1[0] = (int)((2u << 16) | (1u << 20) | (4u << 22) | (1u << 25));
    g1[1] = (int)((tensor_d0 & 0xFFFFu) << 16);          // barrier_addr=0 | dim0 lo16
    g1[2] = (int)((tensor_d0 >> 16) | ((tile_rows & 0xFFFFu) << 16)); // dim0 hi | dim1 lo
    g1[3] = (int)(32u << 16);                            // dim1 hi=0 | tile_dim0=32
    g1[4] = (int)(tile_rows & 0xFFFFu);                  // tile_dim1 | tile_dim2=0
    g1[5] = (int)(unsigned)(row_stride & 0xFFFFFFFFu);   // dim0_stride lo32
    g1[6] = (int)(unsigned)((row_stride >> 32) & 0xFFFFu); // dim0_stride hi | dim1_stride=0
    g1[7] = 0;

    const v4i gz = {0, 0, 0, 0};
#if __clang_major__ >= 23
    const v8i gz8 = {0, 0, 0, 0, 0, 0, 0, 0};
    __builtin_amdgcn_tensor_load_to_lds(g0, g1, gz, gz, gz8, 0);
#else
    __builtin_amdgcn_tensor_load_to_lds(g0, g1, gz, gz, 0);
#endif
}
#endif

__device__ __forceinline__ unsigned lds_off(const void* p) {
    return (unsigned)(unsigned long long)p;   // LDS generic addr: low 32 bits = offset
}

// ---------------------------------------------------------------------------
// GEMM: Out[M x N] = A[M x K] @ W[N x K]^T   (+ optional D[n]*U[m,n] epilogue)
// f32 WMMA 16x16x4. Block: 256 thr = 8 waves (2M x 4N), wave tile 16M x 32N,
// block tile 32M x 128N, K-chunk = 32, double-buffered LDS tiles filled by TDM.
// LDS rows padded to 34 dwords -> conflict-free ds_load_b64 fragment reads.
// ---------------------------------------------------------------------------
#define KC   32
#define LDW  34   // padded row stride in dwords

template <bool FUSE_DU>
__global__ __launch_bounds__(256)
void gemm_wmma_f32(const float* __restrict__ A, const float* __restrict__ W,
                   float* __restrict__ Out, const float* __restrict__ U,
                   const float* __restrict__ D, int M, int N, int K)
{
    __shared__ float As[2][32 * LDW];    //  8.5 KB
    __shared__ float Ws[2][128 * LDW];   // 34.0 KB

    const int lane = threadIdx.x & 31;
    const int wv   = __builtin_amdgcn_readfirstlane((int)(threadIdx.x >> 5));
    const int wm   = wv >> 2;        // 0..1
    const int wn   = wv & 3;         // 0..3
    const int half = lane >> 4;      // K-pair select
    const int l    = lane & 15;

    const int m0 = blockIdx.y * 32;
    const int n0 = blockIdx.x * 128;
    const int nc = K / KC;

#if USE_TDM
    // per-wave DMA slices: 4 rows of A, 16 rows of W per chunk
    const float* gA = A + (size_t)(m0 + wv * 4) * K;
    const float* gW = W + (size_t)(n0 + wv * 16) * K;
    const unsigned ldsA0 = lds_off(&As[0][(wv * 4) * LDW]);
    const unsigned ldsA1 = lds_off(&As[1][(wv * 4) * LDW]);
    const unsigned ldsW0 = lds_off(&Ws[0][(wv * 16) * LDW]);
    const unsigned ldsW1 = lds_off(&Ws[1][(wv * 16) * LDW]);

    // prologue: chunk 0 -> buffer 0
    tdm_load_tile_f32(ldsA0, gA, (unsigned)K, (unsigned long long)K, 4);
    tdm_load_tile_f32(ldsW0, gW, (unsigned)K, (unsigned long long)K, 16);
#else
    const float* arow = A + (size_t)(m0 + wm * 16 + l) * K + 2 * half;
    const float* w0g  = W + (size_t)(n0 + wn * 32 + l) * K + 2 * half;
    const float* w1g  = W + (size_t)(n0 + wn * 32 + 16 + l) * K + 2 * half;
#endif

    v8f acc0 = {0.f, 0.f, 0.f, 0.f, 0.f, 0.f, 0.f, 0.f};
    v8f acc1 = {0.f, 0.f, 0.f, 0.f, 0.f, 0.f, 0.f, 0.f};

    for (int c = 0; c < nc; ++c) {
        const int cur = c & 1;
#if USE_TDM
        // issue next chunk's DMA into the other buffer, then wait for current:
        // TDM ops complete in order per wave, so <=2 outstanding => chunk c done.
        if (c + 1 < nc) {
            const int kn = (c + 1) * KC;
            tdm_load_tile_f32(cur ? ldsA0 : ldsA1, gA + kn,
                              (unsigned)K, (unsigned long long)K, 4);
            tdm_load_tile_f32(cur ? ldsW0 : ldsW1, gW + kn,
                              (unsigned)K, (unsigned long long)K, 16);
            __builtin_amdgcn_s_wait_tensorcnt(2);
        } else {
            __builtin_amdgcn_s_wait_tensorcnt(0);
        }
        __syncthreads();   // chunk c visible to all waves

        const float* as = &As[cur][(wm * 16 + l) * LDW + 2 * half];
        const float* b0s = &Ws[cur][(wn * 32 + l) * LDW + 2 * half];
        const float* b1s = &Ws[cur][(wn * 32 + 16 + l) * LDW + 2 * half];
        #pragma unroll
        for (int k = 0; k < KC; k += 4) {
            v2f a  = *(const v2f*)(as + k);
            v2f b0 = *(const v2f*)(b0s + k);
            v2f b1 = *(const v2f*)(b1s + k);
            acc0 = __builtin_amdgcn_wmma_f32_16x16x4_f32(false, a, false, b0,
                                                         (short)0, acc0, false, false);
            acc1 = __builtin_amdgcn_wmma_f32_16x16x4_f32(false, a, false, b1,
                                                         (short)0, acc1, false, false);
        }
        __syncthreads();   // compute done before this buffer is re-filled
#else
        const int kb = c * KC;
        #pragma unroll
        for (int k = 0; k < KC; k += 4) {
            v2f a  = *(const v2f*)(arow + kb + k);
            v2f b0 = *(const v2f*)(w0g + kb + k);
            v2f b1 = *(const v2f*)(w1g + kb + k);
            acc0 = __builtin_amdgcn_wmma_f32_16x16x4_f32(false, a, false, b0,
                                                         (short)0, acc0, false, false);
            acc1 = __builtin_amdgcn_wmma_f32_16x16x4_f32(false, a, false, b1,
                                                         (short)0, acc1, false, false);
        }
#endif
    }

    // C/D layout: VGPR r -> M = r + 8*half, N = lane&15 (per 16-wide tile)
    #pragma unroll
    for (int r = 0; r < 8; ++r) {
        const int row = m0 + wm * 16 + r + 8 * half;
        const int c0  = n0 + wn * 32 + l;
        const int c1  = n0 + wn * 32 + 16 + l;
        float vo0 = acc0[r];
        float vo1 = acc1[r];
        if (FUSE_DU) {
            vo0 += D[c0] * U[(size_t)row * N + c0];
            vo1 += D[c1] * U[(size_t)row * N + c1];
        }
        Out[(size_t)row * N + c0] = vo0;
        Out[(size_t)row * N + c1] = vo1;
    }
}

// ---------------------------------------------------------------------------
// FDTD recurrence: single persistent workgroup (32 wave32 = 1024 threads) on
// one WGP; state in LDS (3 * 9216 * 4B = 108 KB < 320 KB); 2 barriers/step.
// Reads Bu[t,g] from ws and overwrites it in place with p[t,g].
// ---------------------------------------------------------------------------
#define CPT 9   // cells per thread: 9216 / 1024

__global__ __launch_bounds__(1024)
void fdtd_kernel(const float* __restrict__ c_in, const float* __restrict__ kp_in,
                 const float* __restrict__ k_in, float* __restrict__ bu_p)
{
    extern __shared__ float smem[];
    float* p_s = smem;           // [G]
    float* oxs = smem + G;       // [G]
    float* oys = smem + 2 * G;   // [G]

    const int   tid    = threadIdx.x;
    const float dt     = 3.0f;
    const float dx     = 1.0f;
    const float inv2dx = 1.0f / (2.0f * dx);
    const float max_c  = 0.7f * dx / (dt * sqrtf(2.0f));

    int   nl[CPT], nr[CPT], nu[CPT], nd[CPT];
    float c2dt[CPT], ido[CPT], idp[CPT];
    float pr[CPT], oxr[CPT], oyr[CPT];

    #pragma unroll
    for (int i = 0; i < CPT; ++i) {
        const int cc = tid * CPT + i;
        const int r  = cc / GS;
        const int q  = cc - r * GS;
        nl[i] = r * GS + (q == 0 ? GS - 1 : q - 1);
        nr[i] = r * GS + (q == GS - 1 ? 0 : q + 1);
        nu[i] = (r == 0 ? GS - 1 : r - 1) * GS + q;
        nd[i] = (r == GS - 1 ? 0 : r + 1) * GS + q;

        float cv = c_in[cc];
        cv       = fminf(fmaxf(cv, 0.1f), max_c);
        c2dt[i]  = cv * cv * dt;
        const float kpv = log1pf(__expf(kp_in[cc]));  // softplus
        const float kv  = log1pf(__expf(k_in[cc]));
        ido[i] = 1.0f / (1.0f + dt * kv);
        idp[i] = 1.0f / (1.0f + dt * kpv);

        pr[i] = 0.f; oxr[i] = 0.f; oyr[i] = 0.f;
        p_s[cc] = 0.f;
    }
    __syncthreads();

    for (int t = 0; t < T; ++t) {
        // prefetch next step's forcing slice into cache (global_prefetch_b8)
        if (t + 1 < T)
            __builtin_prefetch(bu_p + (size_t)(t + 1) * G + tid * CPT, 0, 1);

        float bu[CPT];
        #pragma unroll
        for (int i = 0; i < CPT; ++i) {
            const int cc = tid * CPT + i;
            bu[i] = bu_p[(size_t)t * G + cc];
            const float gx = (p_s[nr[i]] - p_s[nl[i]]) * inv2dx;
            const float gy = (p_s[nd[i]] - p_s[nu[i]]) * inv2dx;
            oxr[i] = oxr[i] - dt * gx;   // ox_star
            oyr[i] = oyr[i] - dt * gy;   // oy_star
            oxs[cc] = oxr[i];
            oys[cc] = oyr[i];
        }
        __syncthreads();
        #pragma unroll
        for (int i = 0; i < CPT; ++i) {
            const int cc = tid * CPT + i;
            const float dxx    = (oxs[nr[i]] - oxs[nl[i]]) * inv2dx;
            const float dyy    = (oys[nd[i]] - oys[nu[i]]) * inv2dx;
            const float p_star = pr[i] - c2dt[i] * (dxx + dyy) + dt * bu[i];
            const float pn     = p_star * idp[i];
            pr[i]  = pn;
            oxr[i] *= ido[i];
            oyr[i] *= ido[i];
            p_s[cc] = pn;                      // next step's stencil input
            bu_p[(size_t)t * G + cc] = pn;     // overwrite Bu with p (in place)
        }
        __syncthreads();
    }
}

// ---------------------------------------------------------------------------
extern "C" void kernel_launch(void* const* d_in, const int* in_sizes, int n_in,
                              void* d_out, int out_size, void* d_ws, size_t ws_size,
                              hipStream_t stream)
{
    const float* U  = (const float*)d_in[0];  // (T, H)
    const float* c  = (const float*)d_in[1];  // (G,)
    const float* kp = (const float*)d_in[2];  // (G,)
    const float* k  = (const float*)d_in[3];  // (G,)
    const float* B  = (const float*)d_in[4];  // (G, H)
    const float* C  = (const float*)d_in[5];  // (H, G)
    const float* D  = (const float*)d_in[6];  // (H,)
    float* out  = (float*)d_out;              // (T, H)
    float* bu_p = (float*)d_ws;               // T*G floats (Bu, then p in place)

    (void)hipFuncSetAttribute((const void*)fdtd_kernel,
                              hipFuncAttributeMaxDynamicSharedMemorySize,
                              3 * G * (int)sizeof(float));

    // 1) Bu = U @ B^T       : M=T, N=G, K=H
    {
        dim3 grid(G / 128, T / 32);
        gemm_wmma_f32<false><<<grid, 256, 0, stream>>>(U, B, bu_p,
                                                       nullptr, nullptr, T, G, H);
    }
    // 2) sequential FDTD recurrence (Bu -> p, in place)
    fdtd_kernel<<<1, 1024, 3 * G * sizeof(float), stream>>>(c, kp, k, bu_p);

    // 3) ys = p @ C^T + D*U : M=T, N=H, K=G
    {
        dim3 grid(H / 128, T / 32);
        gemm_wmma_f32<true><<<grid, 256, 0, stream>>>(bu_p, C, out, U, D, T, H, G);
    }
}